// SNN_22282290331764
// MI455X (gfx1250) — compile-verified
//
#include <hip/hip_runtime.h>

// SNN forward, B=500000 rows, T=32 steps, layers 2->2->2.
// Store-bandwidth bound: 3 tensors x [T,B,2] f32 = 384 MB out @ 23.3 TB/s ~= 16.5 us floor.
// Strategy: 1 thread per batch row, all state in VGPRs, cur1 computed once per wave
// via two V_WMMA_F32_16X16X4_F32 (A = fc1_w rows duplicated at M={0,1,8,9} so each
// lane's result lands in its own c[0],c[1]); outputs streamed with non-temporal
// 8-byte coalesced stores (256 B per wave per store instruction).

typedef float v2f __attribute__((ext_vector_type(2)));
typedef float v8f __attribute__((ext_vector_type(8)));

__global__ __launch_bounds__(256) void snn_atan_fwd_kernel(
    const float* __restrict__ x,
    const float* __restrict__ fc1_w, const float* __restrict__ fc1_b,
    const float* __restrict__ beta1, const float* __restrict__ thr1p,
    const float* __restrict__ fc2_w, const float* __restrict__ fc2_b,
    const float* __restrict__ beta2, const float* __restrict__ thr2p,
    float* __restrict__ out, int nbatch, int nsteps)
{
  const int row  = blockIdx.x * blockDim.x + threadIdx.x;
  const int lane = threadIdx.x & 31;
  const int base = row - lane;                 // wave-uniform
  if (base >= nbatch) return;                  // whole-wave early out, EXEC stays full

  // ---- per-lane input row ----
  float2 xv = make_float2(0.f, 0.f);
  if (row < nbatch) xv = ((const float2*)x)[row];

  // ---- layer-1 current: cur1[n][m] = sum_k x[n][k] * fc1_w[m][k] + fc1_b[m] ----
  float cur1_0, cur1_1;
  const bool fullwave = (base + 32 <= nbatch); // wave-uniform
  if (fullwave) {
    // A (16x4 f32, 2 VGPRs; VGPR0: K0 lanes0-15 / K2 lanes16-31; VGPR1: K1/K3):
    // rows {0,1} = fc1_w rows, duplicated at rows {8,9}; K=2,3 columns are zero.
    v2f a; a[0] = 0.f; a[1] = 0.f;
    if (lane < 16) {
      int mm = lane & 7;                       // lane 0,8 -> row0 ; lane 1,9 -> row1
      if (mm < 2) { a[0] = fc1_w[2*mm + 0]; a[1] = fc1_w[2*mm + 1]; }
    }
    // B (4x16 f32, 2 VGPRs; VGPR0: K0 lanes0-15 / K2 lanes16-31; VGPR1: K1/K3):
    // B1 columns = rows base..base+15, B2 columns = rows base+16..base+31.
    float sx = __shfl(xv.x, lane ^ 16, 32);
    float sy = __shfl(xv.y, lane ^ 16, 32);
    v2f b1, b2;
    b1[0] = (lane < 16) ? xv.x : 0.f;
    b1[1] = (lane < 16) ? xv.y : 0.f;
    b2[0] = (lane < 16) ? sx   : 0.f;
    b2[1] = (lane < 16) ? sy   : 0.f;
    v8f cz = {};
    v8f c1 = __builtin_amdgcn_wmma_f32_16x16x4_f32(false, a, false, b1, (short)0, cz, false, false);
    v8f c2 = __builtin_amdgcn_wmma_f32_16x16x4_f32(false, a, false, b2, (short)0, cz, false, false);
    // D layout: VGPR r -> M=r (lanes 0-15, N=lane) / M=r+8 (lanes 16-31, N=lane-16).
    // lanes 0-15: C1[0..1][lane] ; lanes 16-31: C2[8..9][lane-16] (duplicated rows).
    cur1_0 = (lane < 16) ? c1[0] : c2[0];
    cur1_1 = (lane < 16) ? c1[1] : c2[1];
  } else {
    // partial-wave fallback (not hit for B % 32 == 0), keeps WMMA EXEC-all-ones legal
    cur1_0 = fmaf(xv.x, fc1_w[0], xv.y * fc1_w[1]);
    cur1_1 = fmaf(xv.x, fc1_w[2], xv.y * fc1_w[3]);
  }
  cur1_0 += fc1_b[0];
  cur1_1 += fc1_b[1];

  // ---- scalar parameters (uniform, scalar-cached) ----
  const float b1c = fminf(fmaxf(beta1[0], 0.f), 1.f);
  const float b2c = fminf(fmaxf(beta2[0], 0.f), 1.f);
  const float t1  = thr1p[0];
  const float t2  = thr2p[0];
  const float w200 = fc2_w[0], w201 = fc2_w[1];
  const float w210 = fc2_w[2], w211 = fc2_w[3];
  const float c2b0 = fc2_b[0], c2b1 = fc2_b[1];

  // ---- output tensors: [T, B, 2] each, concatenated ----
  const size_t S = (size_t)nsteps * (size_t)nbatch * 2u;  // floats per tensor
  float* o_spk1 = out;
  float* o_spk2 = out + S;
  float* o_mem2 = out + 2 * S;

  float m10 = 0.f, m11 = 0.f, m20 = 0.f, m21 = 0.f;
  const bool active = (row < nbatch);

  for (int t = 0; t < nsteps; ++t) {
    // layer 1: reset from previous mem, leak+integrate, spike
    float r10 = (m10 > t1) ? t1 : 0.f;
    float r11 = (m11 > t1) ? t1 : 0.f;
    m10 = fmaf(b1c, m10, cur1_0) - r10;
    m11 = fmaf(b1c, m11, cur1_1) - r11;
    float s10 = (m10 > t1) ? 1.f : 0.f;
    float s11 = (m11 > t1) ? 1.f : 0.f;
    // layer 2 current: cur2[i] = spk1_0*W2[i][0] + spk1_1*W2[i][1] + b2[i]
    float cu20 = fmaf(s10, w200, fmaf(s11, w201, c2b0));
    float cu21 = fmaf(s10, w210, fmaf(s11, w211, c2b1));
    float r20 = (m20 > t2) ? t2 : 0.f;
    float r21 = (m21 > t2) ? t2 : 0.f;
    m20 = fmaf(b2c, m20, cu20) - r20;
    m21 = fmaf(b2c, m21, cu21) - r21;
    float s20 = (m20 > t2) ? 1.f : 0.f;
    float s21 = (m21 > t2) ? 1.f : 0.f;

    if (active) {
      size_t idx = 2u * ((size_t)t * (size_t)nbatch + (size_t)row);
      v2f v1; v1[0] = s10; v1[1] = s11;
      v2f v2; v2[0] = s20; v2[1] = s21;
      v2f v3; v3[0] = m20; v3[1] = m21;
      __builtin_nontemporal_store(v1, (v2f*)(o_spk1 + idx));
      __builtin_nontemporal_store(v2, (v2f*)(o_spk2 + idx));
      __builtin_nontemporal_store(v3, (v2f*)(o_mem2 + idx));
    }
  }
}

extern "C" void kernel_launch(void* const* d_in, const int* in_sizes, int n_in,
                              void* d_out, int out_size, void* d_ws, size_t ws_size,
                              hipStream_t stream) {
  const float* x     = (const float*)d_in[0];
  const float* fc1_w = (const float*)d_in[1];
  const float* fc1_b = (const float*)d_in[2];
  const float* beta1 = (const float*)d_in[3];
  const float* thr1  = (const float*)d_in[4];
  const float* fc2_w = (const float*)d_in[5];
  const float* fc2_b = (const float*)d_in[6];
  const float* beta2 = (const float*)d_in[7];
  const float* thr2  = (const float*)d_in[8];

  const int nbatch = in_sizes[0] / 2;                 // x is [B, 2]
  const int nsteps = out_size / (nbatch * 6);         // 3 tensors * B * 2 per step

  const int block = 256;                              // 8 wave32 per block
  const int grid  = (nbatch + block - 1) / block;
  snn_atan_fwd_kernel<<<grid, block, 0, stream>>>(
      x, fc1_w, fc1_b, beta1, thr1, fc2_w, fc2_b, beta2, thr2,
      (float*)d_out, nbatch, nsteps);
}